// LoRALayer_66494683676793
// MI455X (gfx1250) — compile-verified
//
#include <hip/hip_runtime.h>
#include <cstdint>

typedef float v2f __attribute__((ext_vector_type(2)));
typedef float v8f __attribute__((ext_vector_type(8)));
typedef unsigned int u32x4 __attribute__((ext_vector_type(4)));
typedef int i32x4 __attribute__((ext_vector_type(4)));
typedef int i32x8 __attribute__((ext_vector_type(8)));

#define M_DIM 256
#define K_DIM 8192
#define N_DIM 8192
#define R_DIM 16
#define KC 32                  // K-chunk staged per iteration
#define NCHUNK (K_DIM / KC)    // 256
#define WAVES 4
#define BLOCK_N (WAVES * 16)   // 64 N-rows per block
#define XP 257                 // x^T pitch in float2 units (pad -> no bank conflicts)

// ---------------------------------------------------------------------------
// Kernel 1: xb[m][r] = sum_k x[m,k] * Bm[r,k]   (256 x 16, tiny)
// ---------------------------------------------------------------------------
__global__ __launch_bounds__(256) void lora_xb(const float* __restrict__ x,
                                               const float* __restrict__ Bm,
                                               float* __restrict__ xb) {
  int t = blockIdx.x * blockDim.x + threadIdx.x;  // 0..4095
  int m = t >> 4;
  int r = t & 15;
  const float4* xr = (const float4*)(x + (size_t)m * K_DIM);
  const float4* br = (const float4*)(Bm + (size_t)r * K_DIM);
  float acc = 0.0f;
  for (int k = 0; k < K_DIM / 4; ++k) {
    float4 xv = xr[k];
    float4 bv = br[k];
    acc += xv.x * bv.x + xv.y * bv.y + xv.z * bv.z + xv.w * bv.w;
  }
  xb[m * R_DIM + r] = acc;
}

// ---------------------------------------------------------------------------
// Kernel 2: out[m][n] = sum_k x[m,k] W[n,k]  +  sum_r xb[m,r] A[n,r]
// Block owns 64 N-rows for ALL of M=256 -> W streamed from HBM exactly once.
// WMMA (computed transposed): A-op = W tile (16n x 4k), B-op = x^T (4k x 16m).
// Both LDS tiles stored as float2-interleaved (k, k+1) so every WMMA operand
// is one contiguous ds_load_b64 into an even-aligned VGPR pair.
// Double-buffered: TDM (W tile) + cooperative x staging for chunk i+1 overlap
// with the 128 WMMAs of chunk i; one barrier per chunk.
// ---------------------------------------------------------------------------
__global__ __launch_bounds__(128) void lora_gemm(const float* __restrict__ x,
                                                 const float* __restrict__ W,
                                                 const float* __restrict__ Amat,
                                                 const float* __restrict__ xb,
                                                 float* __restrict__ out) {
  __shared__ v2f XS[2][(KC / 2) * XP];       // x^T: XS[buf][k/2][m] = {x[m,k], x[m,k+1]}
  __shared__ v2f WS[2][BLOCK_N * (KC / 2)];  // W:   WS[buf][n][k/2] (row-major, TDM fill)
  __shared__ float TB[WAVES][16 * 20];       // per-wave epilogue transpose buffer

  const int tid = threadIdx.x;
  const int wave = tid >> 5;
  const int lane = tid & 31;
  const int r = lane & 15;   // row within 16
  const int hl = lane >> 4;  // half-wave select
  const int nBase = blockIdx.x * BLOCK_N;  // block's global n origin
  const int nw = wave * 16;                // wave's n_local origin
  const v2f* __restrict__ x2 = (const v2f*)x;

  v8f acc[16];
  const v8f vzero = {0.f, 0.f, 0.f, 0.f, 0.f, 0.f, 0.f, 0.f};
#pragma unroll
  for (int i = 0; i < 16; ++i) acc[i] = vzero;

  // ---- staging helpers -----------------------------------------------------
  auto stage_x = [&](int buf, int k0) {
    // Coalesced global float2 reads (lanes 0..15 -> 128B contiguous of one m-row).
    for (int e = tid; e < (KC / 2) * M_DIM; e += 128) {
      int kp = e & (KC / 2 - 1);  // 0..15
      int m = e >> 4;
      XS[buf][kp * XP + m] = x2[(size_t)m * (K_DIM / 2) + (k0 >> 1) + kp];
    }
  };
  auto issue_tdm = [&](int buf, int k0) {
    // TDM: 64 n-rows x 32 K of f32, row stride K, into WS[buf] (wave 0 only).
    unsigned long long ga =
        (unsigned long long)(uintptr_t)(W + (size_t)nBase * K_DIM + k0);
    unsigned ldsOff = (unsigned)(uintptr_t)(void*)&WS[buf][0];
    u32x4 g0;
    g0[0] = 1u;                                    // count=1, user descriptor
    g0[1] = ldsOff;                                // lds_addr (bytes)
    g0[2] = (unsigned)(ga & 0xFFFFFFFFull);        // global_addr[31:0]
    g0[3] = (unsigned)((ga >> 32) & 0x1FFFFFFull)  // global_addr[56:32]
            | (2u << 30);                          // type = 2 ("image")
    i32x8 g1;
    g1[0] = (2 << 16);                             // data_size = 4 bytes
    g1[1] = (int)((K_DIM & 0xFFFF) << 16);         // tensor_dim0 lo16
    g1[2] = (int)(((K_DIM >> 16) & 0xFFFF)         // tensor_dim0 hi16
                  | ((N_DIM & 0xFFFF) << 16));     // tensor_dim1 lo16
    g1[3] = (int)(((N_DIM >> 16) & 0xFFFF)         // tensor_dim1 hi16
                  | (KC << 16));                   // tile_dim0 = 32
    g1[4] = BLOCK_N;                               // tile_dim1 = 64, tile_dim2 = 0
    g1[5] = K_DIM;                                 // tensor_dim0_stride lo32
    g1[6] = 0;
    g1[7] = 0;
    i32x4 gz4 = {0, 0, 0, 0};
    i32x8 gz8 = {0, 0, 0, 0, 0, 0, 0, 0};
    __builtin_amdgcn_tensor_load_to_lds(g0, g1, gz4, gz4, gz8, 0);
  };

  // ---- prologue: fill buffer 0 --------------------------------------------
  stage_x(0, 0);
  if (wave == 0) {
    issue_tdm(0, 0);
    __builtin_amdgcn_s_wait_tensorcnt(0);
  }
  __syncthreads();

  // ---- main pipeline -------------------------------------------------------
  for (int i = 0; i < NCHUNK; ++i) {
    const int cur = i & 1;
    const int nxt = cur ^ 1;
    if (i + 1 < NCHUNK) {
      stage_x(nxt, (i + 1) * KC);         // overlaps with compute below
      if (wave == 0) issue_tdm(nxt, (i + 1) * KC);
    }

    const v2f* __restrict__ xs = &XS[cur][0];
    const v2f* __restrict__ ws = &WS[cur][0];
#pragma unroll
    for (int kk = 0; kk < KC; kk += 4) {
      // A operand: lane r = n row; lanes 0-15 K=kk,kk+1; lanes 16-31 K=kk+2,kk+3
      v2f a = ws[(nw + r) * (KC / 2) + (kk >> 1) + hl];
#pragma unroll
      for (int mc = 0; mc < 16; ++mc) {
        v2f b = xs[((kk >> 1) + hl) * XP + mc * 16 + r];  // one ds_load_b64
        acc[mc] = __builtin_amdgcn_wmma_f32_16x16x4_f32(
            /*neg_a=*/false, a, /*neg_b=*/false, b,
            /*c_mod=*/(short)0, acc[mc], /*reuse_a=*/false, /*reuse_b=*/false);
      }
    }

    if (wave == 0) __builtin_amdgcn_s_wait_tensorcnt(0);
    __syncthreads();
  }

  // ---- epilogue: transpose 16x16 tiles through LDS, fuse rank-16 LoRA ------
  float* tb = &TB[wave][0];
#pragma unroll 1
  for (int mc = 0; mc < 16; ++mc) {
    // D layout: VGPR v, lane -> n_in_tile = v + hl*8, m_in_tile = r
#pragma unroll
    for (int v = 0; v < 8; ++v) tb[r * 20 + (v + hl * 8)] = acc[mc][v];
    // Same-wave LDS ops complete in order; safe to read back without barrier.
#pragma unroll
    for (int i2 = 0; i2 < 2; ++i2) {
      int idx = lane + i2 * 32;  // 0..63 -> 16 rows x 4 float4 segments
      int row = idx >> 2;
      int seg = idx & 3;
      int m = mc * 16 + row;
      float4 val = *(const float4*)&tb[row * 20 + seg * 4];
      int n = nBase + nw + seg * 4;
      float* vp = (float*)&val;
#pragma unroll
      for (int j = 0; j < 4; ++j) {
        float lora = 0.0f;
#pragma unroll
        for (int rr = 0; rr < R_DIM; ++rr)
          lora += xb[m * R_DIM + rr] * Amat[(size_t)(n + j) * R_DIM + rr];
        vp[j] += lora;
      }
      *(float4*)&out[(size_t)m * N_DIM + n] = val;
    }
  }
}

// ---------------------------------------------------------------------------
extern "C" void kernel_launch(void* const* d_in, const int* in_sizes, int n_in,
                              void* d_out, int out_size, void* d_ws, size_t ws_size,
                              hipStream_t stream) {
  const float* x = (const float*)d_in[0];   // [256, 8192]
  const float* W = (const float*)d_in[1];   // [8192, 8192]
  const float* A = (const float*)d_in[2];   // [8192, 16]
  const float* Bm = (const float*)d_in[3];  // [16, 8192]
  float* out = (float*)d_out;               // [256, 8192]
  float* xb = (float*)d_ws;                 // 256*16 floats scratch

  lora_xb<<<16, 256, 0, stream>>>(x, Bm, xb);
  lora_gemm<<<N_DIM / BLOCK_N, 128, 0, stream>>>(x, W, A, xb, out);
}